// NLLMDNLoss_1795296330126
// MI455X (gfx1250) — compile-verified
//
#include <hip/hip_runtime.h>
#include <hip/hip_bf16.h>

// Problem constants (from reference): N=16384, T=60, M=8, K=2
#define N_TOT   16384
#define T_DIM   60
#define M_DIM   8
#define K_DIM   2
#define LOG_2PI 1.8378770664093453f

typedef __attribute__((ext_vector_type(2))) float v2f;
typedef __attribute__((ext_vector_type(4))) float v4f;
typedef __attribute__((ext_vector_type(8))) float v8f;

// ---------------------------------------------------------------------------
// Kernel 0: zero the accumulator slab in workspace (poisoned by harness).
// ws[0..59]   = per-t sum of comp_loss*mask over n
// ws[64..123] = per-t sum of mask over n
// ---------------------------------------------------------------------------
__global__ __launch_bounds__(128) void mdn_init(float* __restrict__ ws) {
    ws[threadIdx.x] = 0.0f;   // 128 threads zero 128 floats
}

// ---------------------------------------------------------------------------
// Kernel 1: streaming GMM NLL. One thread per (n,t), n-major flattening so the
// big mu/sigma streams are read fully coalesced. Non-temporal hints on the
// one-shot 190MB streams; pi (reused 60x per n) uses cached loads.
// ---------------------------------------------------------------------------
__global__ __launch_bounds__(256) void mdn_main(
    const float* __restrict__ mu,     // (N,T,M,K)
    const float* __restrict__ sigma,  // (N,T,M,K,K)
    const float* __restrict__ pi,     // (N,M)
    const float* __restrict__ x,      // (N,T,K)
    const float* __restrict__ mask,   // (N,T)
    float* __restrict__ loss_acc,     // [T]
    float* __restrict__ mask_acc) {   // [T]
    __shared__ float sh[2 * T_DIM];

    const int tid = threadIdx.x;
    if (tid < 2 * T_DIM) sh[tid] = 0.0f;
    __syncthreads();

    const unsigned idx = blockIdx.x * 256u + tid;   // < N*T = 983040 exactly
    const unsigned n   = idx / T_DIM;
    const int      t   = (int)(idx - n * T_DIM);

    // --- streaming loads (read-once -> non-temporal) ---
    const v2f  xv = __builtin_nontemporal_load((const v2f*)(x + (size_t)idx * K_DIM));
    const float mk = __builtin_nontemporal_load(mask + idx);

    // --- log_softmax(pi[n,:]) denominator (cached: heavy reuse across t) ---
    const v4f* pp = (const v4f*)(pi + (size_t)n * M_DIM);
    const v4f pa = pp[0], pb = pp[1];
    float pv[M_DIM] = {pa.x, pa.y, pa.z, pa.w, pb.x, pb.y, pb.z, pb.w};
    float pmax = pv[0];
#pragma unroll
    for (int m = 1; m < M_DIM; ++m) pmax = fmaxf(pmax, pv[m]);
    float ps = 0.0f;
#pragma unroll
    for (int m = 0; m < M_DIM; ++m) ps += expf(pv[m] - pmax);
    const float plse = pmax + logf(ps);   // log_pi[m] = pv[m] - plse

    // --- per-mixture MVN log-prob via closed-form 2x2 ---
    const v4f* sg = (const v4f*)(sigma + (size_t)idx * (M_DIM * K_DIM * K_DIM));
    const v2f* mm = (const v2f*)(mu    + (size_t)idx * (M_DIM * K_DIM));
    float v[M_DIM];
#pragma unroll
    for (int m = 0; m < M_DIM; ++m) {
        const v4f s = __builtin_nontemporal_load(sg + m);   // [a b; b c]
        const v2f u = __builtin_nontemporal_load(mm + m);
        const float d0 = xv.x - u.x;
        const float d1 = xv.y - u.y;
        const float a = s.x, b = s.y, c = s.w;
        const float det  = fmaf(a, c, -(b * b));            // a*c - b^2 (SPD => >0)
        const float quad = fmaf(c * d0, d0, fmaf(a * d1, d1, -2.0f * b * d0 * d1));
        const float maha = quad / det;
        // -0.5*(k*log2pi + maha) - log|L| ;  log|L| = 0.5*log(det)
        const float mvn = fmaf(-0.5f, maha + 2.0f * LOG_2PI, -0.5f * logf(det));
        v[m] = (pv[m] - plse) + mvn;
    }

    // --- logsumexp over mixtures ---
    float vmax = v[0];
#pragma unroll
    for (int m = 1; m < M_DIM; ++m) vmax = fmaxf(vmax, v[m]);
    float vs = 0.0f;
#pragma unroll
    for (int m = 0; m < M_DIM; ++m) vs += expf(v[m] - vmax);
    const float gmm_lp = vmax + logf(vs);

    // --- segmented reduction into per-t bins (LDS atomics, then global) ---
    atomicAdd(&sh[t],          -gmm_lp * mk);
    atomicAdd(&sh[T_DIM + t],  mk);
    __syncthreads();
    if (tid < T_DIM) {
        atomicAdd(&loss_acc[tid], sh[tid]);
        atomicAdd(&mask_acc[tid], sh[T_DIM + tid]);
    }
}

// ---------------------------------------------------------------------------
// Kernel 2: final sum_t loss[t]/npred[t] using V_WMMA_F32_16X16X4_F32.
// A = 16x4 all-ones, B = 4x16 holding the 60 ratios (zero padded). Then
// D[0][j] = column sums of B, and sum_j D[0][j] = sum of ALL B entries,
// which is layout-invariant. One wave, EXEC all ones.
// ---------------------------------------------------------------------------
__global__ __launch_bounds__(32) void mdn_final(const float* __restrict__ loss_acc,
                                                const float* __restrict__ mask_acc,
                                                float* __restrict__ out) {
    const int lane = threadIdx.x;   // 32 lanes == one wave32
    // Each lane feeds two B-matrix slots: value indices lane and lane+32.
    float r0 = 0.0f, r1 = 0.0f;
    if (lane < T_DIM)      r0 = loss_acc[lane]      / mask_acc[lane];
    if (lane + 32 < T_DIM) r1 = loss_acc[lane + 32] / mask_acc[lane + 32];

    v2f a; a.x = 1.0f; a.y = 1.0f;   // ones A (16x4)
    v2f b; b.x = r0;   b.y = r1;     // data B (4x16)
    v8f acc = {};
    acc = __builtin_amdgcn_wmma_f32_16x16x4_f32(
        /*neg_a=*/false, a, /*neg_b=*/false, b,
        /*c_mod=*/(short)0, acc, /*reuse_a=*/false, /*reuse_b=*/false);

    // D row M=0 (column sums) lives in VGPR0 on lanes 0..15.
    float s = acc[0];
    s += __shfl_xor(s, 1, 32);
    s += __shfl_xor(s, 2, 32);
    s += __shfl_xor(s, 4, 32);
    s += __shfl_xor(s, 8, 32);       // lanes 0..15 reduce among themselves
    if (lane == 0) out[0] = s;
}

// ---------------------------------------------------------------------------
extern "C" void kernel_launch(void* const* d_in, const int* in_sizes, int n_in,
                              void* d_out, int out_size, void* d_ws, size_t ws_size,
                              hipStream_t stream) {
    const float* mu    = (const float*)d_in[0];
    const float* sigma = (const float*)d_in[1];
    const float* pi    = (const float*)d_in[2];
    const float* x     = (const float*)d_in[3];
    const float* mask  = (const float*)d_in[4];

    float* ws       = (float*)d_ws;
    float* loss_acc = ws;        // 60 floats
    float* mask_acc = ws + 64;   // 60 floats

    mdn_init<<<1, 128, 0, stream>>>(ws);

    const int total = N_TOT * T_DIM;          // 983040 = 3840 * 256
    mdn_main<<<total / 256, 256, 0, stream>>>(mu, sigma, pi, x, mask,
                                              loss_acc, mask_acc);

    mdn_final<<<1, 32, 0, stream>>>(loss_acc, mask_acc, (float*)d_out);
}